// CrossViewContrastiveLoss_71159018160383
// MI455X (gfx1250) — compile-verified
//
#include <hip/hip_runtime.h>
#include <math.h>

typedef float v2f __attribute__((ext_vector_type(2)));
typedef float v8f __attribute__((ext_vector_type(8)));

// Problem constants (from reference)
constexpr int   K        = 256;        // feature dim -> output is K x K
constexpr int   N_TOTAL  = 131072;     // reduction dim
constexpr int   CHUNK    = 2048;       // n-rows per workgroup
constexpr int   NCHUNK   = N_TOTAL / CHUNK;  // 64
constexpr int   CBLOCKS  = 8;          // 2 (i) x 4 (j) blocks of 128 x 64
constexpr float EPSF     = 2.220446049250313e-16f;
constexpr float AP1      = 10.0f;      // ALPHA + 1

// ---------------------------------------------------------------------------
// Kernel 0: zero the 256x256 fp32 accumulation workspace
// ---------------------------------------------------------------------------
__global__ void zero_ws_kernel(float* __restrict__ Cws) {
  int t = blockIdx.x * blockDim.x + threadIdx.x;
  if (t < K * K) Cws[t] = 0.0f;
}

// ---------------------------------------------------------------------------
// Kernel 1: C += V1^T * V2 over an n-chunk, using V_WMMA_F32_16X16X4_F32.
// Grid: CBLOCKS * NCHUNK workgroups, 256 threads (8 waves) each.
// Each wave computes a 2x2 grid of 16x16 fp32 accumulator tiles.
// ---------------------------------------------------------------------------
__global__ __launch_bounds__(256) void gemm_accum_kernel(
    const float* __restrict__ v1, const float* __restrict__ v2,
    float* __restrict__ Cws) {
  const int cb    = blockIdx.x % CBLOCKS;   // which 128x64 C block
  const int chunk = blockIdx.x / CBLOCKS;   // which n-chunk
  const int ib    = cb >> 2;                // 0..1  -> i block * 128
  const int jb    = cb & 3;                 // 0..3  -> j block * 64
  const int w     = threadIdx.x >> 5;       // wave 0..7
  const int lane  = threadIdx.x & 31;
  const int half  = lane >> 4;              // lane group (0: lanes 0-15, 1: 16-31)
  const int col   = lane & 15;

  // wave tile origin inside the C block: 4 waves along i (32 each), 2 along j
  const int i0 = ib * 128 + (w & 3) * 32;
  const int j0 = jb * 64  + (w >> 2) * 32;

  v8f acc00 = {}, acc01 = {}, acc10 = {}, acc11 = {};

  const int n_begin = chunk * CHUNK;
  // Per ISA layout for 32-bit 16x4 A (and 4x16 B):
  //   VGPR0 holds K = 2*half, VGPR1 holds K = 2*half + 1, M/N = col.
  const float* ap = v1 + (size_t)(n_begin + 2 * half) * K + i0 + col;
  const float* bp = v2 + (size_t)(n_begin + 2 * half) * K + j0 + col;

#pragma unroll 2
  for (int s = 0; s < CHUNK / 4; ++s) {
    v2f a0, a1, b0, b1;
    a0.x = ap[0];   a0.y = ap[K];        // i-tile 0
    a1.x = ap[16];  a1.y = ap[K + 16];   // i-tile 1
    b0.x = bp[0];   b0.y = bp[K];        // j-tile 0
    b1.x = bp[16];  b1.y = bp[K + 16];   // j-tile 1

    acc00 = __builtin_amdgcn_wmma_f32_16x16x4_f32(false, a0, false, b0,
                                                  (short)0, acc00, false, false);
    acc01 = __builtin_amdgcn_wmma_f32_16x16x4_f32(false, a0, false, b1,
                                                  (short)0, acc01, false, false);
    acc10 = __builtin_amdgcn_wmma_f32_16x16x4_f32(false, a1, false, b0,
                                                  (short)0, acc10, false, false);
    acc11 = __builtin_amdgcn_wmma_f32_16x16x4_f32(false, a1, false, b1,
                                                  (short)0, acc11, false, false);
    ap += 4 * K;
    bp += 4 * K;
  }

  // C/D layout: VGPR v holds row M = v + 8*half, column N = col.
  v8f accs[4] = {acc00, acc01, acc10, acc11};
#pragma unroll
  for (int t = 0; t < 4; ++t) {
    const int ti = t >> 1, tj = t & 1;
    const int rbase = i0 + ti * 16 + 8 * half;
    const int cidx  = j0 + tj * 16 + col;
#pragma unroll
    for (int v = 0; v < 8; ++v) {
      __hip_atomic_fetch_add(&Cws[(size_t)(rbase + v) * K + cidx], accs[t][v],
                             __ATOMIC_RELAXED, __HIP_MEMORY_SCOPE_AGENT);
    }
  }
}

// ---------------------------------------------------------------------------
// Kernel 2: symmetrize, normalize, marginals, log-loss -> scalar.
// One workgroup of 256 threads; thread t owns row t of the 256x256 matrix.
// Symmetrized P is symmetric, so row sums == column sums (p_i == p_j^T).
// ---------------------------------------------------------------------------
__global__ __launch_bounds__(256) void finalize_kernel(
    const float* __restrict__ Cws, float* __restrict__ out) {
  __shared__ float rowsum[K];
  __shared__ float part[K];
  const int t = threadIdx.x;

  float rs = 0.0f;
  for (int j = 0; j < K; ++j)
    rs += 0.5f * (Cws[t * K + j] + Cws[j * K + t]);
  rowsum[t] = rs;
  __syncthreads();

  float total = 0.0f;
  for (int j = 0; j < K; ++j) total += rowsum[j];
  const float inv   = 1.0f / total;
  const float pi    = fmaxf(rowsum[t] * inv, EPSF);
  const float lpi   = logf(pi);

  float lp = 0.0f;
  for (int j = 0; j < K; ++j) {
    float pij = fmaxf(0.5f * (Cws[t * K + j] + Cws[j * K + t]) * inv, EPSF);
    float pj  = fmaxf(rowsum[j] * inv, EPSF);
    lp += -pij * (logf(pij) - AP1 * logf(pj) - AP1 * lpi);
  }
  part[t] = lp;
  __syncthreads();

  if (t == 0) {
    float s = 0.0f;
    for (int j = 0; j < K; ++j) s += part[j];
    out[0] = s;
  }
}

// ---------------------------------------------------------------------------
extern "C" void kernel_launch(void* const* d_in, const int* in_sizes, int n_in,
                              void* d_out, int out_size, void* d_ws,
                              size_t ws_size, hipStream_t stream) {
  (void)in_sizes; (void)n_in; (void)out_size; (void)ws_size;
  const float* v1 = (const float*)d_in[0];
  const float* v2 = (const float*)d_in[1];
  float* out = (float*)d_out;
  float* Cws = (float*)d_ws;   // needs K*K*4 = 256 KB of scratch

  zero_ws_kernel<<<(K * K + 255) / 256, 256, 0, stream>>>(Cws);
  gemm_accum_kernel<<<CBLOCKS * NCHUNK, 256, 0, stream>>>(v1, v2, Cws);
  finalize_kernel<<<1, K, 0, stream>>>(Cws, out);
}